// StructuredGaussianMarkovPosterior_90829968376039
// MI455X (gfx1250) — compile-verified
//
#include <hip/hip_runtime.h>

// StructuredGaussianMarkovPosterior — fused single-pass kernel for gfx1250.
//
// Z_t = psi*Z_{t-1} + c_t + sigma*eps_t  evaluated 16 timesteps at a time as
//   Z_block(16t x 16n) = P(16x16 lower-tri psi powers) @ U_block + outer(psiPow, Zcarry)
// via V_WMMA_F32_16X16X4_F32 (4 K-chunks). One wave32 owns 16 chains and all
// S=4 samples (A/L read exactly once). v2: reduced live ranges (base[8] only in
// the load loop, per-sample U built just-in-time), VGPRs capped <256 to kill
// s_set_vgpr_msb churn, and non-temporal output stores (clang ext-vector type).

typedef __attribute__((ext_vector_type(2))) float v2f;
typedef __attribute__((ext_vector_type(4))) float v4f;
typedef __attribute__((ext_vector_type(8))) float v8f;

#define N_  16384
#define T_  256
#define K_  8
#define PD_ 16
#define PS_ 8
#define S_  4

__device__ __forceinline__ float psi_pow(float psi, int e) {
  if (e < 0) return 0.0f;
  float v = 1.0f;
  for (int i = 0; i < e; ++i) v *= psi;
  return v;
}

__device__ __forceinline__ void nt_store4(float* p, float x, float y, float z, float w) {
  v4f v;
  v.x = x; v.y = y; v.z = z; v.w = w;
  __builtin_nontemporal_store(v, (v4f*)p);
}

__global__ __launch_bounds__(256, 4)  // force <=256 VGPRs/wave (1024-VGPR SIMD)
void sgmp_scan_wmma(const float* __restrict__ A, const float* __restrict__ L,
                    const float* __restrict__ V, const float* __restrict__ Y,
                    const float* __restrict__ EPS,
                    const float* __restrict__ p_psi, const float* __restrict__ p_gA,
                    const float* __restrict__ p_gL, const float* __restrict__ p_gV,
                    const float* __restrict__ p_alpha, const float* __restrict__ p_lsig,
                    const float* __restrict__ p_mu, const float* __restrict__ p_lsig0,
                    float* __restrict__ Zout, float* __restrict__ MUout,
                    float* __restrict__ LVout)
{
  const int lane = threadIdx.x & 31;
  const int wave = blockIdx.x * (blockDim.x >> 5) + (threadIdx.x >> 5);
  const int hi   = lane >> 4;        // half-wave (K/M split per ISA operand layouts)
  const int c    = lane & 15;        // chain within tile / matrix column N
  const int n    = wave * 16 + c;    // global chain id

  // ---- scalar params (uniform -> SMEM) ----
  const float psi   = p_psi[0];
  const float alpha = p_alpha[0];
  const float lsig  = p_lsig[0];
  const float lsig0 = p_lsig0[0];
  const float sig   = __expf(lsig);
  const float sig0  = __expf(lsig0);
  const float mu_i  = p_mu[0];
  const float lv1   = 2.0f * lsig;
  const float lv0   = 2.0f * lsig0;

  float gA[K_];
#pragma unroll
  for (int k = 0; k < K_; ++k) gA[k] = p_gA[k];
  float gL[PD_];
#pragma unroll
  for (int p = 0; p < PD_; ++p) gL[p] = p_gL[p];

  // static drive V . gamma_V (both half-waves compute the same chain's value)
  float vterm = 0.0f;
#pragma unroll
  for (int p = 0; p < PS_; ++p) vterm += V[(size_t)n * PS_ + p] * p_gV[p];

  // ---- constant WMMA A-operand: P[m][k] = psi^(m-k) for m>=k else 0 ----
  // f32 A 16x4 layout: lanes 0-15 hold (K0,K1), lanes 16-31 hold (K2,K3); M = lane&15.
  v2f pa[4];
#pragma unroll
  for (int q = 0; q < 4; ++q) {
    const int col = 4 * q + 2 * hi;
    pa[q].x = psi_pow(psi, c - col);
    pa[q].y = psi_pow(psi, c - col - 1);
  }
  // carry column: C[M][n] = psi^(M+1) * Zcarry[n], M = r + 8*hi
  float pcar[8];
#pragma unroll
  for (int r = 0; r < 8; ++r) pcar[r] = psi_pow(psi, r + 8 * hi + 1);

  float Zc[S_] = {0.0f, 0.0f, 0.0f, 0.0f};

  const float* __restrict__ Abase = A + (size_t)n * T_ * K_;
  const float* __restrict__ Lbase = L + (size_t)n * T_ * PD_;
  const float* __restrict__ Ybase = Y + (size_t)n * T_;

  for (int tb = 0; tb < T_; tb += 16) {
    // ---- deterministic drive for this lane's 8 assigned times (B-operand times) ----
    // B 4x16 layout: lanes 0-15 hold rows K={0,1}, lanes 16-31 rows K={2,3} per chunk.
    float base[8];
#pragma unroll
    for (int j = 0; j < 8; ++j) {
      const int t = tb + 4 * (j >> 1) + 2 * hi + (j & 1);
      const float* ap = Abase + (size_t)t * K_;
      const v4f a0 = *(const v4f*)(ap);
      const v4f a1 = *(const v4f*)(ap + 4);
      float b = vterm
          + a0.x * gA[0] + a0.y * gA[1] + a0.z * gA[2] + a0.w * gA[3]
          + a1.x * gA[4] + a1.y * gA[5] + a1.z * gA[6] + a1.w * gA[7];
      const float* lp = Lbase + (size_t)t * PD_;
#pragma unroll
      for (int q4 = 0; q4 < 4; ++q4) {
        const v4f l4 = *(const v4f*)(lp + 4 * q4);
        b += l4.x * gL[4 * q4 + 0] + l4.y * gL[4 * q4 + 1]
           + l4.z * gL[4 * q4 + 2] + l4.w * gL[4 * q4 + 3];
      }
      b += (t > 0) ? (alpha * Ybase[t - 1]) : mu_i;   // t==0: +mu_init, no Y lag
      base[j] = b;
      if (t + 16 < T_) {                               // stream-ahead next block
        __builtin_prefetch(ap + 16 * K_, 0, 0);
        __builtin_prefetch(lp + 16 * PD_, 0, 0);
      }
    }

    // ---- per-sample blocked scan via WMMA (U built just-in-time, short live ranges) ----
#pragma unroll
    for (int s = 0; s < S_; ++s) {
      const float* __restrict__ ep = EPS + (size_t)s * T_ * N_ + n;

      float u[8];
#pragma unroll
      for (int j = 0; j < 8; ++j) {
        const int t = tb + 4 * (j >> 1) + 2 * hi + (j & 1);
        const float se = (t == 0) ? sig0 : sig;
        u[j] = base[j] + se * ep[(size_t)t * N_];
      }

      v8f acc;
#pragma unroll
      for (int r = 0; r < 8; ++r) acc[r] = pcar[r] * Zc[s];
#pragma unroll
      for (int q = 0; q < 4; ++q) {
        v2f b;
        b.x = u[2 * q + 0];
        b.y = u[2 * q + 1];
        acc = __builtin_amdgcn_wmma_f32_16x16x4_f32(
            /*neg_a=*/false, pa[q], /*neg_b=*/false, b,
            /*c_mod=*/(short)0, acc, /*reuse_a=*/false, /*reuse_b=*/false);
      }
      // carry: Z at M=15 lives in acc[7] of lanes 16..31 (N = lane-16)
      Zc[s] = __shfl(acc[7], 16 + c, 32);

      // ---- outputs: this lane holds Z for chain c at t = tb + 8*hi + r ----
      const int    t0      = tb + 8 * hi;
      const size_t outBase = ((size_t)s * N_ + n) * T_ + t0;
      float mu[8];
#pragma unroll
      for (int r = 0; r < 8; ++r) {
        const int t = t0 + r;
        const float e2 = ep[(size_t)t * N_];            // L0-resident reload
        const float se = (t == 0) ? sig0 : sig;
        mu[r] = acc[r] - se * e2;
      }
      const float lva = (t0 == 0) ? lv0 : lv1;          // only t==0 differs
      nt_store4(Zout + outBase,      acc[0], acc[1], acc[2], acc[3]);
      nt_store4(Zout + outBase + 4,  acc[4], acc[5], acc[6], acc[7]);
      nt_store4(MUout + outBase,     mu[0], mu[1], mu[2], mu[3]);
      nt_store4(MUout + outBase + 4, mu[4], mu[5], mu[6], mu[7]);
      nt_store4(LVout + outBase,     lva, lv1, lv1, lv1);
      nt_store4(LVout + outBase + 4, lv1, lv1, lv1, lv1);
    }
  }
}

extern "C" void kernel_launch(void* const* d_in, const int* in_sizes, int n_in,
                              void* d_out, int out_size, void* d_ws, size_t ws_size,
                              hipStream_t stream) {
  (void)in_sizes; (void)n_in; (void)out_size; (void)d_ws; (void)ws_size;
  const float* A     = (const float*)d_in[0];
  const float* L     = (const float*)d_in[1];
  const float* V     = (const float*)d_in[2];
  const float* Y     = (const float*)d_in[3];
  const float* EPS   = (const float*)d_in[4];
  const float* psi   = (const float*)d_in[5];
  const float* gA    = (const float*)d_in[6];
  const float* gLw   = (const float*)d_in[7];
  const float* gVw   = (const float*)d_in[8];
  const float* alpha = (const float*)d_in[9];
  const float* lsig  = (const float*)d_in[10];
  const float* mui   = (const float*)d_in[11];
  const float* lsig0 = (const float*)d_in[12];

  float* Z  = (float*)d_out;
  float* MU = Z  + (size_t)S_ * N_ * T_;
  float* LV = MU + (size_t)S_ * N_ * T_;

  // N/16 = 1024 waves, 8 waves per 256-thread block -> 128 blocks; all waves full.
  sgmp_scan_wmma<<<dim3(N_ / 16 / 8), dim3(256), 0, stream>>>(
      A, L, V, Y, EPS, psi, gA, gLw, gVw, alpha, lsig, mui, lsig0, Z, MU, LV);
}